// BEVConvSV8_19696720019818
// MI455X (gfx1250) — compile-verified
//
#include <hip/hip_runtime.h>
#include <hip/hip_bf16.h>

// ---------------------------------------------------------------------------
// BEV pipeline for MI455X (gfx1250, wave32).
//  1) scatter points -> 5-channel BEV accumulators (atomics)
//  2) finalize bev/avg_z in place
//  3) 3x { implicit-GEMM conv3x3 via V_WMMA_F32_16X16X4_F32 with LDS-staged
//          operands (async global->LDS), fused BN-stats, BN+ReLU+maxpool2 }
// Roofline: tensor traffic ~0.6 GB (stats fused) -> ~30us floor @23.3TB/s.
// Conv FLOPs 14.3 GF fp32 -> WMMA (2048 FLOP/instr) keeps convs off the
// critical path; fp32 WMMA preserves reference numerics for BN statistics.
// ---------------------------------------------------------------------------

#define BEV_B 4
#define BEV_H 1600
#define BEV_W 1408
#define BN_EPS_F 1e-5f

typedef float v2f __attribute__((ext_vector_type(2)));
typedef float v8f __attribute__((ext_vector_type(8)));

#if __has_builtin(__builtin_amdgcn_global_load_async_to_lds_b32)
#define HAVE_ASYNC_LDS 1
typedef __attribute__((address_space(1))) int glb_i32;   // printed as __device__ int*
typedef __attribute__((address_space(3))) int lds_i32;   // printed as __shared__ int*
#else
#define HAVE_ASYNC_LDS 0
#endif

// ---- workspace layout (float element offsets) -----------------------------
// x0 : B*5*H*W       = 45,056,000   (reused for y2: 4*12*800*704 = 27,033,600)
// y  : B*8*H*W       = 72,089,600   (y1; reused for y3: 4*32*400*352 = 18,022,400)
// xp : 4*8*800*704   = 18,022,400   (x1; reused for x2: 4*12*400*352 = 6,758,400)
// st : 64 sums + 64 scale/shift
#define OFF_X0 0ul
#define OFF_Y  45056000ul
#define OFF_XP 117145600ul
#define OFF_ST 135168000ul
#define OFF_SC 135168064ul

// ---------------------------------------------------------------------------
// float atomic min/max via ordered-int encoding (correct across signs)
// ---------------------------------------------------------------------------
__device__ __forceinline__ void atomicMaxF(float* a, float v) {
  if (v >= 0.0f) atomicMax((int*)a, __float_as_int(v));
  else           atomicMin((unsigned int*)a, __float_as_uint(v));
}
__device__ __forceinline__ void atomicMinF(float* a, float v) {
  if (v >= 0.0f) atomicMin((int*)a, __float_as_int(v));
  else           atomicMax((unsigned int*)a, __float_as_uint(v));
}

// ---------------------------------------------------------------------------
// Stage 0: BEV accumulation
// ---------------------------------------------------------------------------
__global__ void init_bev_kernel(float* x0) {
  long idx = (long)blockIdx.x * blockDim.x + threadIdx.x;
  long total = (long)BEV_B * 5 * BEV_H * BEV_W;
  if (idx >= total) return;
  int ch = (int)((idx / ((long)BEV_H * BEV_W)) % 5);
  float v = 0.0f;
  if (ch == 2) v = 10.0f;        // zmin init
  else if (ch == 3) v = -10.0f;  // zmax init
  x0[idx] = v;                   // cnt, zsum, imax init 0
}

__global__ void scatter_kernel(const float* __restrict__ pts, int N,
                               float* __restrict__ x0) {
  int n = blockIdx.x * blockDim.x + threadIdx.x;
  if (n >= N) return;
  const float* p = pts + (long)n * 5;
  int   b  = (int)p[0];
  float px = p[1], py = p[2], pz = p[3], pi = p[4];
  int xp = (int)(px * 20.0f);               // 1408 / 70.4
  int yp = (int)((py + 40.0f) * 20.0f);     // 1600 / 80
  if (xp < 0 || xp >= BEV_W || yp < 0 || yp >= BEV_H) return;
  long plane = (long)BEV_H * BEV_W;
  float* cell = x0 + (long)b * 5 * plane + (long)yp * BEV_W + xp;
  atomicAdd(cell + 0 * plane, 1.0f);   // cnt
  atomicAdd(cell + 1 * plane, pz);     // zsum
  atomicMinF(cell + 2 * plane, pz);    // zmin
  atomicMaxF(cell + 3 * plane, pz);    // zmax
  atomicMaxF(cell + 4 * plane, pi);    // imax (init 0, i>=0)
}

__global__ void finalize_bev_kernel(float* x0) {
  long idx = (long)blockIdx.x * blockDim.x + threadIdx.x;
  long plane = (long)BEV_H * BEV_W;
  long total = (long)BEV_B * plane;
  if (idx >= total) return;
  long b = idx / plane, hw = idx % plane;
  float* cell = x0 + b * 5 * plane + hw;
  float cnt = cell[0];
  float cs  = (cnt == 0.0f) ? 1.0f : cnt;
  cell[1 * plane] = cell[1 * plane] / cs;  // avg_z
  cell[0]         = cs * (1.0f / 50.0f);   // bev
}

// ---------------------------------------------------------------------------
// Conv 3x3 SAME as implicit GEMM; one 16(pix)x16(cout) tile per wave.
// W % 16 == 0 at every stage => each tile lies in ONE image row (no tails).
// A: per-wave LDS slab 3 rows x 18 cols x Cin, fetched global->LDS (async
//    data-mover path when available). B: block-shared [K][16] weight matrix,
//    zero-padded in both K (to mult of 4) and N (to 16).
// K accumulated with V_WMMA_F32_16X16X4_F32, fully unrolled at compile time.
// Per-channel BN sums fused: LDS block reduction + 1 global atomic/channel.
// ---------------------------------------------------------------------------
constexpr int aoff(int k) {  // slab offset for K index k (cin,kh,kw packed)
  return ((k / 9) * 3 + ((k % 9) / 3)) * 18 + (k % 3);
}

template <int Cin, int Cout, int H, int W>
__global__ void __launch_bounds__(256)
conv_wmma_kernel(const float* __restrict__ x, const float* __restrict__ wgt,
                 const float* __restrict__ bias, float* __restrict__ y,
                 float* __restrict__ sums) {
  constexpr int Ktot = Cin * 9;
  constexpr int KPAD = (Ktot + 3) & ~3;        // K padded to wmma step
  constexpr int CINP = (KPAD + 8) / 9;         // slab cin count incl. K pad
  constexpr int SLAB = CINP * 3 * 18;          // floats per wave slab
  constexpr int TPR  = W / 16;                 // tiles per row

  __shared__ float sw[KPAD * 16];              // weights [k][n], zero padded
  __shared__ float slab[8][SLAB];              // per-wave input slab
  __shared__ float sstat[32];                  // per-block channel sums

  const int tid  = threadIdx.x;
  const int lane = tid & 31;
  const int wv   = tid >> 5;
  const int tile = blockIdx.x * 8 + wv;

  // ---- cooperative weight load into LDS
  for (int i = tid; i < KPAD * 16; i += 256) {
    int k = i >> 4, n = i & 15;
    sw[i] = (k < Ktot && n < Cout) ? wgt[n * Ktot + k] : 0.0f;
  }
  if (tid < 32) sstat[tid] = 0.0f;

  // ---- tile coordinates (all divisors compile-time)
  const int trow = tile / TPR;
  const int c0   = (tile - trow * TPR) * 16;
  const int yy   = trow % H;
  const int b    = trow / H;

  // ---- input slab: rows [yy-1, yy+1], cols [c0-1, c0+16], all Cin
  float* sl = slab[wv];
  for (int i = lane; i < SLAB; i += 32) {
    int cc  = i % 18;
    int t2  = i / 18;
    int r   = t2 % 3;
    int cin = t2 / 3;
    int iy  = yy + r - 1;
    int ix  = c0 + cc - 1;
    bool ok = (cin < Cin) && (iy >= 0) && (iy < H) && (ix >= 0) && (ix < W);
#if HAVE_ASYNC_LDS
    if (ok) {
      const float* gp = &x[(((long)b * Cin + cin) * H + iy) * (long)W + ix];
      __builtin_amdgcn_global_load_async_to_lds_b32(
          (glb_i32*)gp, (lds_i32*)(sl + i), 0, 0);
    } else {
      sl[i] = 0.0f;
    }
#else
    sl[i] = ok ? x[(((long)b * Cin + cin) * H + iy) * (long)W + ix] : 0.0f;
#endif
  }
#if HAVE_ASYNC_LDS
#if __has_builtin(__builtin_amdgcn_s_wait_asynccnt)
  __builtin_amdgcn_s_wait_asynccnt(0);
#else
  asm volatile("s_wait_asynccnt 0" ::: "memory");
#endif
#endif
  __syncthreads();

  // ---- WMMA K loop, fully unrolled; offsets fold to constants + cndmask
  const int n  = lane & 15;   // A pixel column == B/C/D column index
  const int hi = lane >> 4;   // lane-half: K pair select (A/B), M+8 (C/D)
  v8f acc = {};
#pragma unroll
  for (int kb = 0; kb < KPAD; kb += 4) {
    v2f av, bv;
    av.x = sl[(hi ? aoff(kb + 2) : aoff(kb + 0)) + n];
    av.y = sl[(hi ? aoff(kb + 3) : aoff(kb + 1)) + n];
    bv.x = sw[(kb + 2 * hi + 0) * 16 + n];
    bv.y = sw[(kb + 2 * hi + 1) * 16 + n];
    acc = __builtin_amdgcn_wmma_f32_16x16x4_f32(false, av, false, bv, (short)0,
                                                acc, false, false);
  }

  // ---- bias, fused BN partial sums, vectorized store
  float vals[8];
  float s = 0.0f, ss = 0.0f;
  const bool nok = (n < Cout);
  const float bs = nok ? bias[n] : 0.0f;
#pragma unroll
  for (int r = 0; r < 8; ++r) {
    float v = acc[r] + bs;
    vals[r] = v;
    s += v;
    ss += v * v;
  }
  if (nok) {
    atomicAdd(&sstat[2 * n + 0], s);
    atomicAdd(&sstat[2 * n + 1], ss);
    float* out = &y[(((long)b * Cout + n) * H + yy) * (long)W + c0 + hi * 8];
    ((float4*)out)[0] = make_float4(vals[0], vals[1], vals[2], vals[3]);
    ((float4*)out)[1] = make_float4(vals[4], vals[5], vals[6], vals[7]);
  }
  __syncthreads();
  if (tid < 2 * Cout) atomicAdd(&sums[tid], sstat[tid]);
}

// ---------------------------------------------------------------------------
// BN finalize + fused BN/ReLU/maxpool2
// ---------------------------------------------------------------------------
__global__ void zero_kernel(float* p, int n) {
  int i = threadIdx.x;
  if (i < n) p[i] = 0.0f;
}

__global__ void bn_finalize_kernel(const float* __restrict__ sums,
                                   const float* __restrict__ g,
                                   const float* __restrict__ be,
                                   float* __restrict__ scsh, int C, float invN) {
  int c = threadIdx.x;
  if (c >= C) return;
  float mean = sums[2 * c] * invN;
  float var  = sums[2 * c + 1] * invN - mean * mean;
  float sc   = g[c] * rsqrtf(var + BN_EPS_F);
  scsh[2 * c]     = sc;
  scsh[2 * c + 1] = be[c] - mean * sc;
}

// out = max(0, max_{2x2}(y*sc+sh))  ==  maxpool(relu(bn(y)))
template <int C, int H, int W>
__global__ void bn_relu_pool_kernel(const float* __restrict__ y,
                                    const float* __restrict__ scsh,
                                    float* __restrict__ out) {
  constexpr int Ho = H / 2, Wo = W / 2;
  constexpr long total = (long)BEV_B * C * Ho * Wo;
  long idx = (long)blockIdx.x * blockDim.x + threadIdx.x;
  if (idx >= total) return;
  int wo = (int)(idx % Wo);
  long t = idx / Wo;
  int ho = (int)(t % Ho); t /= Ho;
  int c  = (int)(t % C);
  int b  = (int)(t / C);
  float sc = scsh[2 * c], sh = scsh[2 * c + 1];
  const float* p = y + (((long)b * C + c) * H + ho * 2) * (long)W + wo * 2;
  float m = 0.0f;
  m = fmaxf(m, p[0] * sc + sh);
  m = fmaxf(m, p[1] * sc + sh);
  m = fmaxf(m, p[W] * sc + sh);
  m = fmaxf(m, p[W + 1] * sc + sh);
  out[idx] = m;
}

// ---------------------------------------------------------------------------
// Host-side stage driver
// ---------------------------------------------------------------------------
static inline int gridFor(long total, int bs) { return (int)((total + bs - 1) / bs); }

template <int Cin, int Cout, int H, int W>
static void run_stage(const float* xin, const float* w, const float* bias,
                      const float* g, const float* be, float* ybuf, float* xout,
                      float* sums, float* scsh, hipStream_t s) {
  constexpr long totalPix = (long)BEV_B * H * W;
  constexpr int nTiles = (int)(totalPix / 16);   // exact: W % 16 == 0
  static_assert(nTiles % 8 == 0, "blocks carry 8 full tiles");
  zero_kernel<<<1, 64, 0, s>>>(sums, 2 * Cout);
  conv_wmma_kernel<Cin, Cout, H, W><<<nTiles / 8, 256, 0, s>>>(xin, w, bias,
                                                               ybuf, sums);
  bn_finalize_kernel<<<1, 32, 0, s>>>(sums, g, be, scsh, Cout,
                                      1.0f / (float)totalPix);
  constexpr long pooled = (long)BEV_B * Cout * (H / 2) * (W / 2);
  bn_relu_pool_kernel<Cout, H, W><<<gridFor(pooled, 256), 256, 0, s>>>(
      ybuf, scsh, xout);
}

extern "C" void kernel_launch(void* const* d_in, const int* in_sizes, int n_in,
                              void* d_out, int out_size, void* d_ws, size_t ws_size,
                              hipStream_t stream) {
  const float* pts = (const float*)d_in[0];
  const float* w1 = (const float*)d_in[1];  const float* b1 = (const float*)d_in[2];
  const float* g1 = (const float*)d_in[3];  const float* be1 = (const float*)d_in[4];
  const float* w2 = (const float*)d_in[5];  const float* b2 = (const float*)d_in[6];
  const float* g2 = (const float*)d_in[7];  const float* be2 = (const float*)d_in[8];
  const float* w3 = (const float*)d_in[9];  const float* b3 = (const float*)d_in[10];
  const float* g3 = (const float*)d_in[11]; const float* be3 = (const float*)d_in[12];
  int N = in_sizes[0] / 5;

  float* ws   = (float*)d_ws;
  float* x0   = ws + OFF_X0;   // BEV accumulators / conv1 input; reused for y2
  float* ybuf = ws + OFF_Y;    // conv1 out; reused for conv3 out
  float* xp   = ws + OFF_XP;   // pooled x1; reused for x2
  float* sums = ws + OFF_ST;
  float* scsh = ws + OFF_SC;

  // Stage 0: BEV scatter + finalize
  long bevTotal = (long)BEV_B * 5 * BEV_H * BEV_W;
  init_bev_kernel<<<gridFor(bevTotal, 256), 256, 0, stream>>>(x0);
  scatter_kernel<<<gridFor((long)N, 256), 256, 0, stream>>>(pts, N, x0);
  finalize_bev_kernel<<<gridFor((long)BEV_B * BEV_H * BEV_W, 256), 256, 0,
                        stream>>>(x0);

  // Stage 1: 5 -> 8 @ 1600x1408, pool -> 800x704
  run_stage<5, 8, BEV_H, BEV_W>(x0, w1, b1, g1, be1, ybuf, xp, sums, scsh,
                                stream);
  // Stage 2: 8 -> 12 @ 800x704, pool -> 400x352   (y2 aliases x0 region)
  run_stage<8, 12, 800, 704>(xp, w2, b2, g2, be2, x0, xp, sums, scsh, stream);
  // Stage 3: 12 -> 32 @ 400x352, pool -> 200x176 into d_out (y3 aliases ybuf)
  run_stage<12, 32, 400, 352>(xp, w3, b3, g3, be3, ybuf, (float*)d_out, sums,
                              scsh, stream);
}